// TopKPTailFreeSampler_52836687676095
// MI455X (gfx1250) — compile-verified
//
#include <hip/hip_runtime.h>
#include <cstdint>

// ---------------- configuration ----------------
#define VOCAB   50257
#define TOPK    50
#define BS      512          // 16 wave32 waves per workgroup
#define CHUNK   8192         // floats per TDM chunk (32 KB)
#define NCH     ((VOCAB + CHUNK - 1) / CHUNK)   // 7 chunks
#define CAP     256          // candidate buffer slots

// dynamic LDS layout (bytes)
constexpr int OFF_BUF0  = 0;
constexpr int OFF_BUF1  = OFF_BUF0 + CHUNK * 4;     // 32768
constexpr int OFF_HIST  = OFF_BUF1 + CHUNK * 4;     // 65536
constexpr int OFF_MRED  = OFF_HIST + 4096 * 4;      // 81920
constexpr int OFF_SRED  = OFF_MRED + BS * 4;        // 83968
constexpr int OFF_CANDX = OFF_SRED + BS * 4;        // 86016
constexpr int OFF_CANDI = OFF_CANDX + CAP * 4;      // 87040
constexpr int OFF_CTRL  = OFF_CANDI + CAP * 4;      // 88064
constexpr int SMEM_BYTES = OFF_CTRL + 64;           // 88128 (< 160 KB even in CU mode)

struct Ctrl {
  float M, Z, cutoff;
  unsigned uMin;
  int n;        // candidate count (atomic)
  int kc;       // kept count after top-k (incl. ties)
  int refine;
  int b1;
  int kA;       // strictly-above-bin count
};

// order-preserving float -> uint key (larger float => larger key)
__device__ __forceinline__ unsigned fkey(float x) {
  unsigned b = __float_as_uint(x);
  return (b & 0x80000000u) ? ~b : (b | 0x80000000u);
}

// online softmax accumulator update
__device__ __forceinline__ void online_upd(float v, float& m, float& s) {
  if (v > m) { s = s * expf(m - v) + 1.0f; m = v; }
  else       { s += expf(v - m); }
}

// ---- JAX threefry2x32 gumbel, key(42) = [0, 42], counts = iota split in halves ----
__device__ float gumbel_at(unsigned j, unsigned H) {
  unsigned x0, x1; int sel;
  if (j < H) { x0 = j;     x1 = j + H; sel = 0; }
  else       { x0 = j - H; x1 = j;     sel = 1; }
  const unsigned ks0 = 0u, ks1 = 42u, ks2 = 0x1BD11BDAu ^ ks0 ^ ks1;
  const unsigned ksv[3] = { ks0, ks1, ks2 };
  const int rotA[4] = {13, 15, 26, 6}, rotB[4] = {17, 29, 16, 24};
  x0 += ks0; x1 += ks1;
#pragma unroll
  for (int i = 0; i < 5; ++i) {
    const int* r = (i & 1) ? rotB : rotA;
#pragma unroll
    for (int q = 0; q < 4; ++q) {
      x0 += x1;
      x1 = (x1 << r[q]) | (x1 >> (32 - r[q]));
      x1 ^= x0;
    }
    x0 += ksv[(i + 1) % 3];
    x1 += ksv[(i + 2) % 3] + (unsigned)(i + 1);
  }
  unsigned bits = sel ? x1 : x0;
  float f = __uint_as_float((bits >> 9) | 0x3F800000u) - 1.0f;   // [0,1)
  const float TINY = 1.17549435e-38f;
  float u = f + TINY;              // floats*(1-tiny)+tiny folds to f+tiny in f32
  u = fmaxf(u, TINY);
  return -logf(-logf(u));
}

// ---------------- CDNA5 Tensor Data Mover staging ----------------
#if defined(__HIP_DEVICE_COMPILE__) && __has_builtin(__builtin_amdgcn_tensor_load_to_lds) && __has_builtin(__builtin_amdgcn_s_wait_tensorcnt)
#define USE_TDM 1
#else
#define USE_TDM 0
#endif

#if !USE_TDM && defined(__HIP_DEVICE_COMPILE__)
#warning "CDNA5_PROBE: tensor_load_to_lds builtin unavailable - compiled fallback global-load path"
#endif

#if USE_TDM
typedef unsigned int u32x4 __attribute__((ext_vector_type(4)));
typedef int          i32x4 __attribute__((ext_vector_type(4)));
typedef int          i32x8 __attribute__((ext_vector_type(8)));

// 1-D tile load: nelem f32 elements from gaddr -> LDS byte offset lds_off
__device__ __forceinline__ void tdm_load_1d(const void* gaddr, unsigned lds_off, int nelem) {
  unsigned long long ga = (unsigned long long)(uintptr_t)gaddr;
  u32x4 g0;
  g0[0] = 1u;                                           // count=1, user mode
  g0[1] = lds_off;                                      // lds_addr (bytes)
  g0[2] = (unsigned)ga;                                 // global_addr[31:0]
  g0[3] = (unsigned)((ga >> 32) & 0x01FFFFFFull) | (2u << 30);  // addr[56:32] | type=2
  i32x8 g1;
  g1[0] = (int)(2u << 16);                              // data_size = 4B
  g1[1] = (int)(((unsigned)nelem & 0xFFFFu) << 16);     // tensor_dim0[15:0]
  g1[2] = (int)((((unsigned)nelem >> 16) & 0xFFFFu) | (1u << 16)); // dim0 hi | tensor_dim1=1 lo
  g1[3] = (int)(((unsigned)nelem & 0xFFFFu) << 16);     // tensor_dim1 hi=0 | tile_dim0
  g1[4] = 1;                                            // tile_dim1 = 1, tile_dim2 = 0
  g1[5] = nelem;                                        // tensor_dim0_stride lo
  g1[6] = 0; g1[7] = 0;
  i32x4 gz = {0, 0, 0, 0};
#if defined(__clang_major__) && (__clang_major__ >= 23)
  i32x8 gz8 = {0, 0, 0, 0, 0, 0, 0, 0};
  __builtin_amdgcn_tensor_load_to_lds(g0, g1, gz, gz, gz8, 0);
#else
  __builtin_amdgcn_tensor_load_to_lds(g0, g1, gz, gz, 0);
#endif
}
#endif

__global__ __launch_bounds__(BS)
void sampler_kernel(const float* __restrict__ X, int* __restrict__ out, unsigned Hhalf) {
  extern __shared__ __align__(16) unsigned char smem[];
  float*    buf0  = (float*)(smem + OFF_BUF0);
  float*    buf1  = (float*)(smem + OFF_BUF1);
  unsigned* hist  = (unsigned*)(smem + OFF_HIST);
  float*    mred  = (float*)(smem + OFF_MRED);
  float*    sred  = (float*)(smem + OFF_SRED);
  float*    candX = (float*)(smem + OFF_CANDX);
  int*      candI = (int*)(smem + OFF_CANDI);
  Ctrl*     ctrl  = (Ctrl*)(smem + OFF_CTRL);

  const int tid = threadIdx.x;
  const int row = blockIdx.x;
  const float* xrow = X + (size_t)row * VOCAB;

  // clear level-1 histogram
  for (int i = tid; i < 4096; i += BS) hist[i] = 0u;
  __syncthreads();

  // -------- pass A: online softmax stats + 12-bit key histogram --------
  float m = -INFINITY, s = 0.0f;

#if USE_TDM
  if (tid == 0) {
    int n0 = (VOCAB < CHUNK) ? VOCAB : CHUNK;
    tdm_load_1d(xrow, (unsigned)(uintptr_t)(const void*)buf0, n0);
  }
  for (int c = 0; c < NCH; ++c) {
    if (tid == 0) {
      int nxt = c + 1;
      if (nxt < NCH) {
        int ne = VOCAB - nxt * CHUNK; if (ne > CHUNK) ne = CHUNK;
        tdm_load_1d(xrow + nxt * CHUNK,
                    (unsigned)(uintptr_t)(const void*)((nxt & 1) ? buf1 : buf0), ne);
        __builtin_amdgcn_s_wait_tensorcnt(1);   // chunk c complete (in-order)
      } else {
        __builtin_amdgcn_s_wait_tensorcnt(0);
      }
    }
    __syncthreads();
    const float* buf = (c & 1) ? buf1 : buf0;
    int cnt = VOCAB - c * CHUNK; if (cnt > CHUNK) cnt = CHUNK;
    // vectorized LDS scan (ds_load_b128)
    int cntv = cnt >> 2;
    const float4* b4 = (const float4*)buf;
    for (int i = tid; i < cntv; i += BS) {
      float4 v = b4[i];
      online_upd(v.x, m, s); atomicAdd(&hist[fkey(v.x) >> 20], 1u);
      online_upd(v.y, m, s); atomicAdd(&hist[fkey(v.y) >> 20], 1u);
      online_upd(v.z, m, s); atomicAdd(&hist[fkey(v.z) >> 20], 1u);
      online_upd(v.w, m, s); atomicAdd(&hist[fkey(v.w) >> 20], 1u);
    }
    for (int i = (cntv << 2) + tid; i < cnt; i += BS) {
      float v = buf[i];
      online_upd(v, m, s);
      atomicAdd(&hist[fkey(v) >> 20], 1u);
    }
    __syncthreads();   // buffer may be overwritten next iteration
  }
#else
  for (int i = tid; i < VOCAB; i += BS) {
    float v = xrow[i];
    online_upd(v, m, s);
    atomicAdd(&hist[fkey(v) >> 20], 1u);
  }
  __syncthreads();
#endif

  // block reduce of (m, s) online pairs
  mred[tid] = m; sred[tid] = s;
  __syncthreads();
  for (int off = BS / 2; off > 0; off >>= 1) {
    if (tid < off) {
      float M0 = mred[tid], S0 = sred[tid];
      float m2 = mred[tid + off], s2 = sred[tid + off];
      if (m2 > M0)            { S0 = S0 * expf(M0 - m2) + s2; M0 = m2; }
      else if (m2 != -INFINITY) S0 += s2 * expf(m2 - M0);
      mred[tid] = M0; sred[tid] = S0;
    }
    __syncthreads();
  }
  if (tid == 0) {
    ctrl->M = mred[0];
    ctrl->Z = sred[0];
    // level-1 scan: bin containing the 50th-largest key
    int acc = 0, b1 = 4095;
    for (int b = 4095; b >= 0; --b) {
      int cb = (int)hist[b];
      if (acc + cb >= TOPK) { b1 = b; break; }
      acc += cb;
    }
    ctrl->b1 = b1;
    ctrl->kA = acc;
    ctrl->refine = (acc + (int)hist[b1] > CAP) ? 1 : 0;
    ctrl->uMin = ((unsigned)b1) << 20;
    ctrl->n = 0;
  }
  __syncthreads();

  // -------- rare refinement: 12-bit second-level histogram --------
  if (ctrl->refine) {
    int b1 = ctrl->b1;
    for (int i = tid; i < 4096; i += BS) hist[i] = 0u;
    __syncthreads();
    for (int i = tid; i < VOCAB; i += BS) {
      unsigned u = fkey(xrow[i]);
      if ((int)(u >> 20) == b1) atomicAdd(&hist[(u >> 8) & 0xFFFu], 1u);
    }
    __syncthreads();
    if (tid == 0) {
      int K = TOPK - ctrl->kA;
      int acc = 0, b2 = 4095;
      for (int b = 4095; b >= 0; --b) {
        int cb = (int)hist[b];
        if (acc + cb >= K) { b2 = b; break; }
        acc += cb;
      }
      ctrl->uMin = ((((unsigned)b1) << 12) | (unsigned)b2) << 8;
      ctrl->n = 0;
    }
    __syncthreads();
  }

  // -------- collect candidates (L2-hot re-read, b128-vectorized) --------
  {
    unsigned uMin = ctrl->uMin;
    // alignment head: row base is only 4B-aligned (VOCAB odd)
    uintptr_t a = (uintptr_t)xrow;
    int head = (int)(((16u - (unsigned)(a & 15u)) & 15u) >> 2);
    if (head > VOCAB) head = VOCAB;
    for (int i = tid; i < head; i += BS) {
      float v = xrow[i];
      if (fkey(v) >= uMin) {
        int pos = atomicAdd(&ctrl->n, 1);
        if (pos < CAP) { candX[pos] = v; candI[pos] = i; }
      }
    }
    const float4* x4 = (const float4*)(xrow + head);
    int nv = (VOCAB - head) >> 2;
    for (int i = tid; i < nv; i += BS) {
      __builtin_prefetch((const void*)(x4 + i + BS), 0, 0);   // global_prefetch_b8
      float4 v = x4[i];
      int g = head + (i << 2);
      if (fkey(v.x) >= uMin) { int p = atomicAdd(&ctrl->n, 1); if (p < CAP) { candX[p] = v.x; candI[p] = g;     } }
      if (fkey(v.y) >= uMin) { int p = atomicAdd(&ctrl->n, 1); if (p < CAP) { candX[p] = v.y; candI[p] = g + 1; } }
      if (fkey(v.z) >= uMin) { int p = atomicAdd(&ctrl->n, 1); if (p < CAP) { candX[p] = v.z; candI[p] = g + 2; } }
      if (fkey(v.w) >= uMin) { int p = atomicAdd(&ctrl->n, 1); if (p < CAP) { candX[p] = v.w; candI[p] = g + 3; } }
    }
    for (int i = head + (nv << 2) + tid; i < VOCAB; i += BS) {
      float v = xrow[i];
      if (fkey(v) >= uMin) {
        int pos = atomicAdd(&ctrl->n, 1);
        if (pos < CAP) { candX[pos] = v; candI[pos] = i; }
      }
    }
  }
  __syncthreads();
  int n = ctrl->n; if (n > CAP) n = CAP;
  for (int i = tid; i < CAP; i += BS)
    if (i >= n) { candX[i] = -INFINITY; candI[i] = 0x7FFFFFFF; }
  __syncthreads();

  // -------- bitonic sort CAP slots, descending by value --------
  for (int k = 2; k <= CAP; k <<= 1) {
    for (int j = k >> 1; j > 0; j >>= 1) {
      if (tid < CAP) {
        int p = tid ^ j;
        if (p > tid) {
          float a = candX[tid], b = candX[p];
          bool up = ((tid & k) == 0);
          if (up ? (a < b) : (a > b)) {
            candX[tid] = b; candX[p] = a;
            int t2 = candI[tid]; candI[tid] = candI[p]; candI[p] = t2;
          }
        }
      }
      __syncthreads();
    }
  }

  // -------- top-k ties + nucleus cutoff (sequential, matches jnp semantics) --------
  if (tid == 0) {
    int kb = TOPK - 1; if (kb > n - 1) kb = n - 1; if (kb < 0) kb = 0;
    float xk = candX[kb];
    int kc = kb + 1;
    while (kc < n && candX[kc] == xk) ++kc;       // keep ties at the k-th value
    float M = ctrl->M, Z = ctrl->Z;
    float cum = 0.0f;
    float cutoff = expf(candX[0] - M) / Z;        // argmax(all-False) == 0 case
    for (int i = 0; i < kc; ++i) {
      float p = expf(candX[i] - M) / Z;
      cum += p;
      if (cum > 0.9f) { cutoff = p; break; }
    }
    ctrl->cutoff = cutoff;
    ctrl->kc = kc;
  }
  __syncthreads();

  // -------- score kept candidates with exact JAX gumbel; argmax (first-index tie) --------
  {
    float M = ctrl->M, Z = ctrl->Z, cutoff = ctrl->cutoff;
    int kc = ctrl->kc;
    float score = -INFINITY; int sidx = 0x7FFFFFFF;
    if (tid < kc) {
      float p = expf(candX[tid] - M) / Z;
      if (p >= cutoff) {
        unsigned flat = (unsigned)row * (unsigned)VOCAB + (unsigned)candI[tid];
        float g = gumbel_at(flat, Hhalf);
        score = logf(powf(p, 1.25f)) + g;          // log(p^(1/0.8)) + gumbel
        sidx = candI[tid];
      }
    }
    sred[tid] = score;
    ((int*)mred)[tid] = sidx;
    __syncthreads();
    for (int off = BS / 2; off > 0; off >>= 1) {
      if (tid < off) {
        float a = sred[tid],       b = sred[tid + off];
        int   ia = ((int*)mred)[tid], ib = ((int*)mred)[tid + off];
        if (b > a || (b == a && ib < ia)) { sred[tid] = b; ((int*)mred)[tid] = ib; }
      }
      __syncthreads();
    }
    if (tid == 0) out[row] = ((int*)mred)[0];
  }
}

extern "C" void kernel_launch(void* const* d_in, const int* in_sizes, int n_in,
                              void* d_out, int out_size, void* d_ws, size_t ws_size,
                              hipStream_t stream) {
  (void)d_ws; (void)ws_size; (void)n_in; (void)in_sizes;
  const float* X = (const float*)d_in[0];
  int* out = (int*)d_out;
  const int B = out_size;   // one sampled index per row
  unsigned long long total = (unsigned long long)B * (unsigned long long)VOCAB;
  unsigned Hhalf = (unsigned)((total + 1ull) / 2ull);
  hipFuncSetAttribute(reinterpret_cast<const void*>(sampler_kernel),
                      hipFuncAttributeMaxDynamicSharedMemorySize, (int)SMEM_BYTES);
  sampler_kernel<<<B, BS, SMEM_BYTES, stream>>>(X, out, Hhalf);
}